// SMPL_91293824844011
// MI455X (gfx1250) — compile-verified
//
#include <hip/hip_runtime.h>
#include <hip/hip_bf16.h>

// ---------------------------------------------------------------------------
// SMPL forward for MI455X (gfx1250, wave32).
// Heavy math (shape+pose blendshape GEMM) runs on V_WMMA_F32_16X16X4_F32 with
// a double-buffered LDS B-panel pipeline; posed vertices + skinning matrices
// stay in LDS (never hit HBM). All hot-loop loads are branch-free (clamped
// indices instead of EXEC-predicated loads).
// ---------------------------------------------------------------------------

#define SMPL_B   512
#define SMPL_V   6890
#define SMPL_C3  (SMPL_V * 3)       // 20670 flattened columns
#define SMPL_NJ  24
#define SMPL_NB  10
#define SMPL_NP  207
#define SMPL_NC  19

#define KCHUNK   16                 // K rows per staged LDS panel
#define NCHUNK   13                 // ceil(207 / 16)
#define BPADW    196                // 192 cols + 4 pad floats per panel row

typedef __attribute__((ext_vector_type(2))) float v2f;
typedef __attribute__((ext_vector_type(8))) float v8f;

#if defined(__gfx1250__) && __has_builtin(__builtin_amdgcn_wmma_f32_16x16x4_f32)
#define HAVE_WMMA_F32X4 1
#else
#define HAVE_WMMA_F32X4 0
#endif

__constant__ int c_parents[SMPL_NJ] = {0, 0, 0, 0, 1, 2, 3, 4, 5, 6, 7, 8,
                                       9, 9, 9, 12, 13, 14, 16, 17, 18, 19, 20, 21};

// ---------------------------------------------------------------------------
// Kernel A: regress the shape basis through the SMPL joint regressor once.
// SJ[k][j][c], k=0..9 -> shape_disp basis, k=10 -> template mesh term.
// ---------------------------------------------------------------------------
__global__ void smpl_joint_basis_kernel(const float* __restrict__ tmpl,
                                        const float* __restrict__ shape_disp,
                                        const float* __restrict__ sjr,
                                        float* __restrict__ SJ) {
  const int o = blockIdx.x;              // 0 .. 11*72-1
  const int k = o / 72;
  const int jc = o % 72;
  const int j = jc / 3;
  const int c = jc % 3;
  float acc = 0.f;
  for (int v = threadIdx.x; v < SMPL_V; v += 256) {
    const float src = (k < SMPL_NB) ? shape_disp[(size_t)k * SMPL_C3 + v * 3 + c]
                                    : tmpl[v * 3 + c];
    acc += src * sjr[v * SMPL_NJ + j];
  }
  __shared__ float red[256];
  red[threadIdx.x] = acc;
  __syncthreads();
  for (int s = 128; s > 0; s >>= 1) {
    if (threadIdx.x < s) red[threadIdx.x] += red[threadIdx.x + s];
    __syncthreads();
  }
  if (threadIdx.x == 0) SJ[o] = red[0];
}

// ---------------------------------------------------------------------------
// Kernel B: Rodrigues per (b, joint). Writes Rs (output #3) and pose_feat (ws).
// ---------------------------------------------------------------------------
__global__ void smpl_rodrigues_kernel(const float* __restrict__ thetas,
                                      float* __restrict__ Rs_out,
                                      float* __restrict__ pose_feat) {
  const int idx = blockIdx.x * blockDim.x + threadIdx.x;
  if (idx >= SMPL_B * SMPL_NJ) return;
  const int b = idx / SMPL_NJ;
  const int j = idx % SMPL_NJ;
  const float r0 = thetas[b * (SMPL_NJ * 3) + j * 3 + 0];
  const float r1 = thetas[b * (SMPL_NJ * 3) + j * 3 + 1];
  const float r2 = thetas[b * (SMPL_NJ * 3) + j * 3 + 2];
  const float ex = r0 + 1e-8f, ey = r1 + 1e-8f, ez = r2 + 1e-8f;
  const float ang = sqrtf(ex * ex + ey * ey + ez * ez);
  const float inv = 1.0f / ang;
  const float ax = r0 * inv, ay = r1 * inv, az = r2 * inv;
  const float c = cosf(ang), s = sinf(ang), t = 1.0f - c;
  float R[9];
  R[0] = c + t * ax * ax;       R[1] = -s * az + t * ax * ay; R[2] = s * ay + t * ax * az;
  R[3] = s * az + t * ay * ax;  R[4] = c + t * ay * ay;       R[5] = -s * ax + t * ay * az;
  R[6] = -s * ay + t * az * ax; R[7] = s * ax + t * az * ay;  R[8] = c + t * az * az;
  float* ro = Rs_out + (size_t)b * (SMPL_NJ * 9) + j * 9;
#pragma unroll
  for (int e = 0; e < 9; ++e) ro[e] = R[e];
  if (j >= 1) {
    float* pf = pose_feat + (size_t)b * SMPL_NP + (j - 1) * 9;
#pragma unroll
    for (int e = 0; e < 9; ++e) pf[e] = R[e] - ((e % 4 == 0) ? 1.0f : 0.0f);
  }
}

// ---------------------------------------------------------------------------
// Kernel C: joints[b,j,c] = SJ_template + sum_k betas[b,k] * SJ[k,j,c].
// ---------------------------------------------------------------------------
__global__ void smpl_joints_kernel(const float* __restrict__ betas,
                                   const float* __restrict__ SJ,
                                   float* __restrict__ joints) {
  const int idx = blockIdx.x * blockDim.x + threadIdx.x;
  if (idx >= SMPL_B * 72) return;
  const int b = idx / 72;
  const int jc = idx % 72;
  float acc = SJ[SMPL_NB * 72 + jc];
#pragma unroll
  for (int k = 0; k < SMPL_NB; ++k) acc += betas[b * SMPL_NB + k] * SJ[k * 72 + jc];
  joints[idx] = acc;
}

// ---------------------------------------------------------------------------
// Kernel D: kinematic chain. One thread per batch. Stores relative skinning
// transforms A[b][24][12] (3x4 rows; homogeneous bottom row implied).
// ---------------------------------------------------------------------------
__global__ void smpl_chain_kernel(const float* __restrict__ Rs,
                                  const float* __restrict__ joints,
                                  float* __restrict__ Aout) {
  const int b = blockIdx.x * blockDim.x + threadIdx.x;
  if (b >= SMPL_B) return;
  const float* Rb = Rs + (size_t)b * (SMPL_NJ * 9);
  const float* Jb = joints + (size_t)b * (SMPL_NJ * 3);
  float* Ab = Aout + (size_t)b * (SMPL_NJ * 12);
#pragma unroll
  for (int r = 0; r < 3; ++r) {
    Ab[r * 4 + 0] = Rb[r * 3 + 0];
    Ab[r * 4 + 1] = Rb[r * 3 + 1];
    Ab[r * 4 + 2] = Rb[r * 3 + 2];
    Ab[r * 4 + 3] = Jb[r];
  }
  for (int i = 1; i < SMPL_NJ; ++i) {
    const int p = c_parents[i];
    float Gp[12];
#pragma unroll
    for (int e = 0; e < 12; ++e) Gp[e] = Ab[p * 12 + e];
    const float* Lr = Rb + i * 9;
    float lt[3];
#pragma unroll
    for (int r = 0; r < 3; ++r) lt[r] = Jb[i * 3 + r] - Jb[p * 3 + r];
    float G[12];
#pragma unroll
    for (int r = 0; r < 3; ++r) {
#pragma unroll
      for (int cc = 0; cc < 3; ++cc) {
        G[r * 4 + cc] = Gp[r * 4 + 0] * Lr[0 * 3 + cc] +
                        Gp[r * 4 + 1] * Lr[1 * 3 + cc] +
                        Gp[r * 4 + 2] * Lr[2 * 3 + cc];
      }
      G[r * 4 + 3] = Gp[r * 4 + 0] * lt[0] + Gp[r * 4 + 1] * lt[1] +
                     Gp[r * 4 + 2] * lt[2] + Gp[r * 4 + 3];
    }
#pragma unroll
    for (int e = 0; e < 12; ++e) Ab[i * 12 + e] = G[e];
  }
  for (int i = 0; i < SMPL_NJ; ++i) {
    const float jx = Jb[i * 3 + 0], jy = Jb[i * 3 + 1], jz = Jb[i * 3 + 2];
#pragma unroll
    for (int r = 0; r < 3; ++r) {
      float* g = Ab + i * 12 + r * 4;
      g[3] = g[3] - (g[0] * jx + g[1] * jy + g[2] * jz);
    }
  }
}

// ---------------------------------------------------------------------------
// Kernel E: fused blendshape GEMM + LBS skinning.
// Block = 128 threads (4 waves). Tile = 16 batches x 64 vertices (192 cols).
// Pose GEMM is double-buffered: global b64 loads of a 16x192 pose_dirs panel
// land in registers while the previous panel feeds 12 WMMAs from LDS.
// All hot loads are unconditional (clamped indices); out-of-range columns
// belong to vertices >= V that are never stored, and out-of-range K rows hit
// zero-padded A fragments (0 * garbage == 0).
// WMMA fragment layout (ISA 7.12.2, wave32):
//   A 16x4 f32: lanes 0-15 hold K0/K1, lanes 16-31 hold K2/K3, row = lane&15
//   B 4x16 f32: mirrored; C/D: VGPR i <-> row i / i+8 per lane-half
// ---------------------------------------------------------------------------
__global__ void __launch_bounds__(128)
smpl_vertex_kernel(const float* __restrict__ betas,
                   const float* __restrict__ tmpl,
                   const float* __restrict__ shape_disp,
                   const float* __restrict__ pose_dirs,
                   const float* __restrict__ pose_feat,
                   const float* __restrict__ lbs,
                   const float* __restrict__ Amat,
                   float* __restrict__ vout) {
  __shared__ float s_beta[16][12];             // betas tile, K padded 10 -> 12
  __shared__ float s_pf[16][208];              // pose_feat tile, K padded 207 -> 208
  __shared__ float s_A[16][SMPL_NJ * 12];
  __shared__ float s_lbs[64][SMPL_NJ];
  // s_B (pose pass) and s_posed (skinning) are disjoint in time -> alias them.
  __shared__ __align__(16) char s_mix[2 * KCHUNK * BPADW * sizeof(float)];
  float(*s_B)[KCHUNK][BPADW] = (float(*)[KCHUNK][BPADW])s_mix;  // [2][16][196]
  float(*s_posed)[192] = (float(*)[192])s_mix;                  // [16][192]

  const int tid = threadIdx.x;
  const int wave = tid >> 5;
  const int lane = tid & 31;
  const int half = lane >> 4;
  const int l16 = lane & 15;

  const int v0 = blockIdx.x * 64;        // first vertex of block
  const int b0 = blockIdx.y * 16;        // first batch of tile
  const int c0 = v0 * 3;                 // first flat column (multiple of 192)

  // ---- prefetch pose_dirs chunk 0 into registers (in flight during staging)
  // 16 rows x 96 float2 = 1536 slots / 128 threads = 12 per thread.
  float2 reg[12];
#pragma unroll
  for (int i = 0; i < 12; ++i) {
    const int s = tid + i * 128;
    const int r = s / 96, q = s % 96;
    size_t flat = (size_t)min(r, SMPL_NP - 1) * SMPL_C3 + c0 + q * 2;
    if (flat > (size_t)(SMPL_NP * SMPL_C3 - 2)) flat = (size_t)(SMPL_NP * SMPL_C3 - 2);
    reg[i] = *(const float2*)(pose_dirs + flat);
  }

  // ---- cooperative staging of per-tile operands ----
  for (int i = tid; i < 16 * 12; i += 128) {
    const int m = i / 12, k = i % 12;
    s_beta[m][k] = (k < SMPL_NB) ? betas[(b0 + m) * SMPL_NB + k] : 0.f;
  }
  for (int i = tid; i < 16 * 208; i += 128) {
    const int m = i / 208, k = i % 208;
    s_pf[m][k] = (k < SMPL_NP) ? pose_feat[(size_t)(b0 + m) * SMPL_NP + k] : 0.f;
  }
  for (int i = tid; i < 16 * SMPL_NJ * 12; i += 128) {
    const int m = i / (SMPL_NJ * 12), e = i % (SMPL_NJ * 12);
    s_A[m][e] = Amat[(size_t)(b0 + m) * (SMPL_NJ * 12) + e];
  }
  for (int i = tid; i < 64 * SMPL_NJ; i += 128) {
    const int vl = i / SMPL_NJ, j = i % SMPL_NJ;
    s_lbs[vl][j] = lbs[min(v0 + vl, SMPL_V - 1) * SMPL_NJ + j];
  }
  __syncthreads();

  int colg[3];
#pragma unroll
  for (int t = 0; t < 3; ++t) colg[t] = c0 + wave * 48 + t * 16 + l16;
  int colc[3];
#pragma unroll
  for (int t = 0; t < 3; ++t) colc[t] = min(colg[t], SMPL_C3 - 1);

  const v8f vzero = {0.f, 0.f, 0.f, 0.f, 0.f, 0.f, 0.f, 0.f};
  v8f acc[3] = {vzero, vzero, vzero};

  // ---- shape pass: K = 10 (padded 12), branch-free direct loads ----
#pragma unroll
  for (int kb = 0; kb < 12; kb += 4) {
    const int r0 = kb + half * 2;
    v2f a;
    a.x = s_beta[l16][r0];                 // rows 10,11 are zero
    a.y = s_beta[l16][r0 + 1];
    const int r0c = min(r0, SMPL_NB - 1);
    const int r1c = min(r0 + 1, SMPL_NB - 1);
#pragma unroll
    for (int t = 0; t < 3; ++t) {
      v2f bf;
      bf.x = shape_disp[(size_t)r0c * SMPL_C3 + colc[t]];
      bf.y = shape_disp[(size_t)r1c * SMPL_C3 + colc[t]];
#if HAVE_WMMA_F32X4
      acc[t] = __builtin_amdgcn_wmma_f32_16x16x4_f32(false, a, false, bf,
                                                     (short)0, acc[t], false, false);
#else
#pragma unroll
      for (int i = 0; i < 8; ++i) acc[t][i] += a.x * bf.x + a.y * bf.y;
#endif
    }
  }

  // ---- pose pass: K = 207, double-buffered 16-row LDS panels ----
  for (int kc = 0; kc < NCHUNK; ++kc) {
    // commit prefetched chunk kc to LDS
#pragma unroll
    for (int i = 0; i < 12; ++i) {
      const int s = tid + i * 128;
      const int r = s / 96, q = s % 96;
      *(float2*)&s_B[kc & 1][r][q * 2] = reg[i];
    }
    __syncthreads();
    // launch global loads for chunk kc+1 (in flight during the WMMAs below)
    if (kc + 1 < NCHUNK) {
#pragma unroll
      for (int i = 0; i < 12; ++i) {
        const int s = tid + i * 128;
        const int r = s / 96, q = s % 96;
        size_t flat = (size_t)min((kc + 1) * KCHUNK + r, SMPL_NP - 1) * SMPL_C3 +
                      c0 + q * 2;
        if (flat > (size_t)(SMPL_NP * SMPL_C3 - 2))
          flat = (size_t)(SMPL_NP * SMPL_C3 - 2);
        reg[i] = *(const float2*)(pose_dirs + flat);
      }
    }
    // 4 K-steps x 3 column tiles = 12 WMMAs from the committed panel
    const float(*Bc)[BPADW] = s_B[kc & 1];
#pragma unroll
    for (int kb = 0; kb < KCHUNK; kb += 4) {
      const int rl = kb + half * 2;                  // local row in panel
      const int rg = kc * KCHUNK + rl;               // global K row
      v2f a;
      a.x = s_pf[l16][min(rg, SMPL_NP)];             // s_pf[207] == 0
      a.y = s_pf[l16][min(rg + 1, SMPL_NP)];
#pragma unroll
      for (int t = 0; t < 3; ++t) {
        const int lc = wave * 48 + t * 16 + l16;
        v2f bf;
        bf.x = Bc[rl][lc];
        bf.y = Bc[rl + 1][lc];
#if HAVE_WMMA_F32X4
        acc[t] = __builtin_amdgcn_wmma_f32_16x16x4_f32(false, a, false, bf,
                                                       (short)0, acc[t], false, false);
#else
#pragma unroll
        for (int i = 0; i < 8; ++i) acc[t][i] += a.x * bf.x + a.y * bf.y;
#endif
      }
    }
    __syncthreads();
  }

  // ---- add template, park posed tile in LDS (aliases s_B; all reads done) ----
#pragma unroll
  for (int t = 0; t < 3; ++t) {
    const int lc = wave * 48 + t * 16 + l16;
    const float tm = tmpl[colc[t]];
#pragma unroll
    for (int i = 0; i < 8; ++i) s_posed[i + half * 8][lc] = acc[t][i] + tm;
  }
  __syncthreads();

  // ---- LBS skinning: 16 batches x 64 vertices = 1024 (b,v) pairs ----
  for (int p = tid; p < 16 * 64; p += 128) {
    const int m = p >> 6;          // batch within tile
    const int vl = p & 63;         // vertex within block
    const int v = v0 + vl;
    float T[12];
#pragma unroll
    for (int e = 0; e < 12; ++e) T[e] = 0.f;
#pragma unroll 4
    for (int j = 0; j < SMPL_NJ; ++j) {
      const float w = s_lbs[vl][j];
      const float* aj = &s_A[m][j * 12];
#pragma unroll
      for (int e = 0; e < 12; ++e) T[e] += w * aj[e];
    }
    const float x = s_posed[m][vl * 3 + 0];
    const float y = s_posed[m][vl * 3 + 1];
    const float z = s_posed[m][vl * 3 + 2];
    if (v < SMPL_V) {
      float* o = vout + ((size_t)(b0 + m) * SMPL_V + v) * 3;
      o[0] = T[0] * x + T[1] * y + T[2] * z + T[3];
      o[1] = T[4] * x + T[5] * y + T[6] * z + T[7];
      o[2] = T[8] * x + T[9] * y + T[10] * z + T[11];
    }
  }
}

// ---------------------------------------------------------------------------
// Kernel F: out_joints[b,j,:] = sum_v vertices[b,v,:] * jr[v,j]
// ---------------------------------------------------------------------------
__global__ void smpl_outjoints_kernel(const float* __restrict__ verts,
                                      const float* __restrict__ jr,
                                      float* __restrict__ out_joints) {
  const int j = blockIdx.x;   // 0..18
  const int b = blockIdx.y;   // 0..511
  float a0 = 0.f, a1 = 0.f, a2 = 0.f;
  const float* vb = verts + (size_t)b * SMPL_V * 3;
  for (int v = threadIdx.x; v < SMPL_V; v += 256) {
    const float w = jr[v * SMPL_NC + j];
    a0 += vb[v * 3 + 0] * w;
    a1 += vb[v * 3 + 1] * w;
    a2 += vb[v * 3 + 2] * w;
  }
  __shared__ float red[256][3];
  red[threadIdx.x][0] = a0;
  red[threadIdx.x][1] = a1;
  red[threadIdx.x][2] = a2;
  __syncthreads();
  for (int s = 128; s > 0; s >>= 1) {
    if (threadIdx.x < s) {
      red[threadIdx.x][0] += red[threadIdx.x + s][0];
      red[threadIdx.x][1] += red[threadIdx.x + s][1];
      red[threadIdx.x][2] += red[threadIdx.x + s][2];
    }
    __syncthreads();
  }
  if (threadIdx.x == 0) {
    float* o = out_joints + ((size_t)b * SMPL_NC + j) * 3;
    o[0] = red[0][0];
    o[1] = red[0][1];
    o[2] = red[0][2];
  }
}

// ---------------------------------------------------------------------------
extern "C" void kernel_launch(void* const* d_in, const int* in_sizes, int n_in,
                              void* d_out, int out_size, void* d_ws, size_t ws_size,
                              hipStream_t stream) {
  (void)in_sizes; (void)n_in; (void)out_size; (void)ws_size;
  const float* betas = (const float*)d_in[0];
  const float* thetas = (const float*)d_in[1];
  const float* tmpl = (const float*)d_in[2];
  const float* shape_disp = (const float*)d_in[3];
  const float* pose_dirs = (const float*)d_in[4];
  const float* sjr = (const float*)d_in[5];
  const float* lbs = (const float*)d_in[6];
  const float* jr = (const float*)d_in[7];

  // output layout: vertices | out_joints | Rs
  float* vertices = (float*)d_out;
  float* out_joints = vertices + (size_t)SMPL_B * SMPL_V * 3;
  float* Rs_out = out_joints + (size_t)SMPL_B * SMPL_NC * 3;

  // workspace layout
  float* ws = (float*)d_ws;
  float* pose_feat = ws;                                 // B * 207
  float* joints = pose_feat + (size_t)SMPL_B * SMPL_NP;  // B * 72
  float* SJ = joints + (size_t)SMPL_B * 72;              // 11 * 72
  float* Amat = SJ + 11 * 72;                            // B * 24 * 12

  smpl_joint_basis_kernel<<<11 * 72, 256, 0, stream>>>(tmpl, shape_disp, sjr, SJ);
  smpl_rodrigues_kernel<<<(SMPL_B * SMPL_NJ + 255) / 256, 256, 0, stream>>>(
      thetas, Rs_out, pose_feat);
  smpl_joints_kernel<<<(SMPL_B * 72 + 255) / 256, 256, 0, stream>>>(betas, SJ, joints);
  smpl_chain_kernel<<<(SMPL_B + 63) / 64, 64, 0, stream>>>(Rs_out, joints, Amat);
  dim3 gridE((SMPL_V + 63) / 64, SMPL_B / 16);
  smpl_vertex_kernel<<<gridE, 128, 0, stream>>>(betas, tmpl, shape_disp, pose_dirs,
                                                pose_feat, lbs, Amat, vertices);
  dim3 gridF(SMPL_NC, SMPL_B);
  smpl_outjoints_kernel<<<gridF, 256, 0, stream>>>(vertices, jr, out_joints);
}